// AdaptiveTopKLoss_51685636440246
// MI455X (gfx1250) — compile-verified
//
#include <hip/hip_runtime.h>
#include <hip/hip_bf16.h>
#include <cstdint>

#define VOCAB        32000
#define NROWS        4096
#define CHUNK_FLOATS 1280      // 5120 bytes = 10 async b128 instructions per wave
#define CHUNK_BYTES  5120
#define NCHUNK       25        // 25 * 1280 = 32000
#define WPB          4         // waves (rows) per block
#define REC_STRIDE   32        // floats per row record in workspace
#define LOG_EPS      -23.025850929940457f   // ln(1e-10)

typedef __attribute__((ext_vector_type(2))) float v2f;
typedef __attribute__((ext_vector_type(8))) float v8f;

// ---------------------------------------------------------------------------
// Kernel A: one wave per row. Streams the row through LDS with async-to-LDS
// double buffering (ASYNCcnt), computing: row max + lse (online softmax),
// sum(x), rank of target value (count of elements ahead of it, jax tie rule),
// and the row's top-20 values (per-lane sorted list + 32-way wave merge).
// ---------------------------------------------------------------------------
__global__ __launch_bounds__(WPB * 32) void atl_stream_stats(
    const float* __restrict__ logits, const int* __restrict__ targets,
    float* __restrict__ rec)
{
  __shared__ __align__(16) float lbuf[WPB][2][CHUNK_FLOATS];
  const int lane = threadIdx.x & 31;
  const int wid  = threadIdx.x >> 5;
  const int row  = blockIdx.x * WPB + wid;

  const float* rowp = logits + (size_t)row * VOCAB;
  const int    tgt  = targets[row];
  const float  x_t  = rowp[tgt];

  // issue one chunk (10 x global_load_async_to_lds_b128, 512B each)
  auto issue = [&](int c, int b) {
    uint32_t la0 = (uint32_t)(uintptr_t)(&lbuf[wid][b][0]) + (uint32_t)(lane * 16);
    uint64_t ga0 = (uint64_t)(uintptr_t)rowp + (uint64_t)c * CHUNK_BYTES
                 + (uint64_t)(lane * 16);
#pragma unroll
    for (int i = 0; i < 10; ++i) {
      uint32_t la = la0 + (uint32_t)(i * 512);
      uint64_t ga = ga0 + (uint64_t)(i * 512);
      asm volatile("global_load_async_to_lds_b128 %0, %1, off"
                   :: "v"(la), "v"(ga) : "memory");
    }
  };

  float m = -INFINITY, s = 0.0f, sumx = 0.0f;
  int   cnt = 0;
  float t[20];
#pragma unroll
  for (int i = 0; i < 20; ++i) t[i] = -INFINITY;

  issue(0, 0);
  issue(1, 1);

  for (int c = 0; c < NCHUNK; ++c) {
    if (c < NCHUNK - 1) { asm volatile("s_wait_asynccnt 10" ::: "memory"); }
    else                { asm volatile("s_wait_asynccnt 0"  ::: "memory"); }

    const float* buf = &lbuf[wid][c & 1][0];
    float4 d[10];
#pragma unroll
    for (int i = 0; i < 10; ++i)
      d[i] = *(const float4*)(buf + i * 128 + lane * 4);

    // chunk-local max then one merged online-softmax update (1 exp/element)
    float cm = -INFINITY;
#pragma unroll
    for (int i = 0; i < 10; ++i) {
      cm = fmaxf(cm, fmaxf(fmaxf(d[i].x, d[i].y), fmaxf(d[i].z, d[i].w)));
    }
    float cs = 0.0f;
#pragma unroll
    for (int i = 0; i < 10; ++i) {
      cs += expf(d[i].x - cm); cs += expf(d[i].y - cm);
      cs += expf(d[i].z - cm); cs += expf(d[i].w - cm);
    }
    {
      float nm = fmaxf(m, cm);
      s = s * expf(m - nm) + cs * expf(cm - nm);
      m = nm;
    }

    // per-element: sum, rank count (jax tie rule), guarded top-20 insert
#pragma unroll
    for (int i = 0; i < 10; ++i) {
      const int colbase = c * CHUNK_FLOATS + i * 128 + lane * 4;
      float xs[4] = {d[i].x, d[i].y, d[i].z, d[i].w};
#pragma unroll
      for (int k = 0; k < 4; ++k) {
        float x = xs[k];
        sumx += x;
        if (x > x_t)                                   cnt++;
        else if (x == x_t && (colbase + k) < tgt)      cnt++;
        if (x > t[19]) {
#pragma unroll
          for (int j = 19; j > 0; --j)
            t[j] = (x > t[j - 1]) ? t[j - 1] : ((x > t[j]) ? x : t[j]);
          t[0] = (x > t[0]) ? x : t[0];
        }
      }
    }

    if (c + 2 < NCHUNK) {
      asm volatile("s_wait_dscnt 0" ::: "memory");  // LDS reads done before refill
      issue(c + 2, c & 1);
    }
  }

  // wave32 butterfly reductions
#pragma unroll
  for (int off = 16; off >= 1; off >>= 1) {
    float om = __shfl_xor(m, off, 32);
    float os = __shfl_xor(s, off, 32);
    float nm = fmaxf(m, om);
    s = s * expf(m - nm) + os * expf(om - nm);
    m = nm;
    sumx += __shfl_xor(sumx, off, 32);
    cnt  += __shfl_xor(cnt,  off, 32);
  }

  // 32-way merge of per-lane sorted top-20 lists -> row top-20
  float* tl = &lbuf[wid][0][0];           // reuse buffer 0 (all async complete)
#pragma unroll
  for (int i = 0; i < 20; ++i) tl[lane * 20 + i] = t[i];

  float* rrow = rec + (size_t)row * REC_STRIDE;
  int ptr = 0;
  for (int k = 0; k < 20; ++k) {
    float v = (ptr < 20) ? tl[lane * 20 + ptr] : -INFINITY;
    int who = lane;
#pragma unroll
    for (int off = 16; off >= 1; off >>= 1) {
      float ov = __shfl_xor(v,   off, 32);
      int   ow = __shfl_xor(who, off, 32);
      if (ov > v || (ov == v && ow < who)) { v = ov; who = ow; }
    }
    if (lane == 0)  rrow[k] = v;
    if (lane == who) ptr++;
  }
  if (lane == 0) {
    rrow[20] = m + logf(s);        // lse over full row
    rrow[21] = sumx;
    rrow[22] = x_t;
    rrow[23] = (float)cnt;         // rank of target (< 20 => in top-20)
  }
}

// ---------------------------------------------------------------------------
// Kernel B: one thread per row. Runs the 20-layer Cauchy-relaxed odd-even
// sort, propagating only the P-column of the ground-truth element, plus the
// top-20 softmax term and the label-smoothed CE terms.
// ---------------------------------------------------------------------------
__global__ __launch_bounds__(256) void atl_sort_ce(
    const float* __restrict__ rec,
    float* __restrict__ topk_terms, float* __restrict__ ce_terms)
{
  const int r = blockIdx.x * 256 + threadIdx.x;
  const float* rr = rec + (size_t)r * REC_STRIDE;

  float tv[20];
#pragma unroll
  for (int i = 0; i < 20; ++i) tv[i] = rr[i];
  const float lse  = rr[20];
  const float sumx = rr[21];
  const float xt   = rr[22];
  const int   rank = (int)rr[23];

  // label-smoothed CE:  nll = lse - x_t ; smooth = lse - mean(x)
  const float nll    = lse - xt;
  const float smooth = lse - sumx * (1.0f / (float)VOCAB);
  ce_terms[r] = 0.95f * nll + 0.05f * smooth;

  float term;
  if (rank >= 20) {
    term = LOG_EPS;   // gt not in top-20: all probs clip to EPS, sum(P_K)=1
  } else {
    // softmax over top vals (tv[0] is the max), pick rank component
    float denom = 0.0f, num = 0.0f;
#pragma unroll
    for (int j = 0; j < 20; ++j) {
      float e = expf(tv[j] - tv[0]);
      denom += e;
      num   += (j == rank) ? e : 0.0f;
    }
    const float p0 = num / denom;

    float v[20], q[20];
#pragma unroll
    for (int i = 0; i < 20; ++i) { v[i] = tv[i]; q[i] = (i == rank) ? 1.0f : 0.0f; }

#pragma unroll
    for (int layer = 0; layer < 20; ++layer) {
      const int off = layer & 1;
#pragma unroll
      for (int p = 0; p + off + 1 < 20; p += 2) {
        const int a = p + off, b = a + 1;
        const float av = v[a], bv = v[b];
        const float alpha = atanf(2.0f * (bv - av)) * 0.31830988618379067f + 0.5f;
        const float ia = 1.0f - alpha;
        v[a] = alpha * av + ia * bv;
        v[b] = alpha * bv + ia * av;
        const float aq = q[a], bq = q[b];
        q[a] = alpha * aq + ia * bq;
        q[b] = alpha * bq + ia * aq;
      }
    }

    const float pg1 = q[19] + q[18];
    const float pg2 = pg1 + q[17];
    const float pg3 = pg2 + q[16];
    const float pg4 = pg3 + q[15];
    auto lc = [](float p) { return logf(fminf(fmaxf(p, 1e-10f), 1.0f)); };
    term = 0.4f * lc(p0) + 0.1f * lc(pg1) + 0.1f * lc(pg2)
         + 0.1f * lc(pg3) + 0.3f * lc(pg4);
  }
  topk_terms[r] = term;
}

// ---------------------------------------------------------------------------
// Kernel C: WMMA-based reduction of the two 4096-element term arrays.
// A = 64 partials per step, B = ones  =>  D rows accumulate sums in the
// matrix unit (v_wmma_f32_16x16x4_f32). sum(D)/16 == array sum.
// ---------------------------------------------------------------------------
__global__ __launch_bounds__(64) void atl_finalize(
    const float* __restrict__ topk_terms, const float* __restrict__ ce_terms,
    const int* __restrict__ epoch, const int* __restrict__ max_epochs,
    float* __restrict__ out)
{
  const int lane = threadIdx.x & 31;
  const int wave = threadIdx.x >> 5;
  const float* arr = (wave == 0) ? topk_terms : ce_terms;

  v8f acc = {};
  v2f ones; ones[0] = 1.0f; ones[1] = 1.0f;
  for (int base = 0; base < NROWS; base += 64) {
    v2f a;
    a[0] = arr[base + lane];
    a[1] = arr[base + 32 + lane];
    acc = __builtin_amdgcn_wmma_f32_16x16x4_f32(
        /*neg_a=*/false, a, /*neg_b=*/false, ones,
        /*c_mod=*/(short)0, acc, /*reuse_a=*/false, /*reuse_b=*/false);
  }
  float p = acc[0] + acc[1] + acc[2] + acc[3] + acc[4] + acc[5] + acc[6] + acc[7];
#pragma unroll
  for (int off = 16; off >= 1; off >>= 1) p += __shfl_xor(p, off, 32);
  p *= (1.0f / 16.0f);   // 16 identical columns in D

  __shared__ float sums[2];
  if (lane == 0) sums[wave] = p;
  __syncthreads();

  if (threadIdx.x == 0) {
    const float topk_loss = -sums[0] / (float)NROWS;
    const float ce_loss   =  sums[1] / (float)NROWS;
    const float ratio  = (float)epoch[0] / (float)max_epochs[0];
    const float topk_w = fmaxf(0.3f, 1.0f - ratio * 0.7f);
    const float ce_w   = 1.0f - topk_w;
    out[0] = topk_w * topk_loss + ce_w * ce_loss;
    out[1] = topk_loss;
    out[2] = ce_loss;
  }
}

// ---------------------------------------------------------------------------
extern "C" void kernel_launch(void* const* d_in, const int* in_sizes, int n_in,
                              void* d_out, int out_size, void* d_ws, size_t ws_size,
                              hipStream_t stream) {
  const float* logits     = (const float*)d_in[0];
  const int*   targets    = (const int*)d_in[1];
  const int*   epoch      = (const int*)d_in[2];
  const int*   max_epochs = (const int*)d_in[3];
  float*       out        = (float*)d_out;

  float* rec        = (float*)d_ws;                    // 4096 * 32 floats
  float* topk_terms = rec + (size_t)NROWS * REC_STRIDE; // 4096 floats
  float* ce_terms   = topk_terms + NROWS;               // 4096 floats

  atl_stream_stats<<<NROWS / WPB, WPB * 32, 0, stream>>>(logits, targets, rec);
  atl_sort_ce<<<NROWS / 256, 256, 0, stream>>>(rec, topk_terms, ce_terms);
  atl_finalize<<<1, 64, 0, stream>>>(topk_terms, ce_terms, epoch, max_epochs, out);
}